// SupCG_3118146257545
// MI455X (gfx1250) — compile-verified
//
#include <hip/hip_runtime.h>
#include <hip/hip_bf16.h>

typedef float v2f __attribute__((ext_vector_type(2)));
typedef float v8f __attribute__((ext_vector_type(8)));

// ---------------------------------------------------------------------------
// WMMA GEMM: C[M x N] = A[M x K] @ B[K x N], optional fused ReLU.
// One wave32 per 16x16 output tile, V_WMMA_F32_16X16X4_F32, K stepped by 4.
// f32 A-frag layout: lanes 0-15 hold M=0..15 {K=k,k+1}; lanes 16-31 {K=k+2,k+3}.
// f32 C/D layout: lane<16 -> (M=v, N=lane); lane>=16 -> (M=v+8, N=lane-16).
// ---------------------------------------------------------------------------
template <int K, int N, bool RELU>
__global__ __launch_bounds__(256) void gemm_wmma_f32_kernel(
    const float* __restrict__ A, const float* __restrict__ B,
    float* __restrict__ C, int M) {
  constexpr int NT = N / 16;
  const int wave = blockIdx.x * (blockDim.x >> 5) + (threadIdx.x >> 5);
  const int mt = wave / NT;
  if (mt * 16 >= M) return;  // wave-uniform exit: EXEC stays all-ones for WMMA
  const int nt = wave % NT;
  const int lane = threadIdx.x & 31;
  const int half = lane >> 4;   // 0: K=k..k+1 | 1: K=k+2..k+3
  const int l = lane & 15;
  const int m0 = mt * 16;
  const int n0 = nt * 16;

  // Rows >= M only pollute unstored D rows; clamp to stay in-bounds.
  const int arow = (m0 + l) < M ? (m0 + l) : 0;
  const float* __restrict__ Ap = A + (size_t)arow * K;

  v8f acc = {};
#pragma unroll 4
  for (int k = 0; k < K; k += 4) {
    v2f a = *(const v2f*)(Ap + k + 2 * half);        // contiguous K pair
    v2f b;
    b.x = B[(size_t)(k + 2 * half) * N + n0 + l];
    b.y = B[(size_t)(k + 2 * half + 1) * N + n0 + l];
    acc = __builtin_amdgcn_wmma_f32_16x16x4_f32(
        /*neg_a=*/false, a, /*neg_b=*/false, b,
        /*c_mod=*/(short)0, acc, /*reuse_a=*/false, /*reuse_b=*/false);
  }

  if (m0 + 16 <= M) {
    // Full tile (always true for M % 16 == 0): wave-uniform, no EXEC games.
    float* __restrict__ Cp = C + (size_t)(m0 + 8 * half) * N + n0 + l;
#pragma unroll
    for (int v = 0; v < 8; ++v) {
      float val = acc[v];
      if (RELU) val = fmaxf(val, 0.0f);
      Cp[(size_t)v * N] = val;
    }
  } else {
    // Ragged final tile: per-row guards.
#pragma unroll
    for (int v = 0; v < 8; ++v) {
      const int orow = m0 + v + 8 * half;
      if (orow < M) {
        float val = acc[v];
        if (RELU) val = fmaxf(val, 0.0f);
        C[(size_t)orow * N + n0 + l] = val;
      }
    }
  }
}

// ---------------------------------------------------------------------------
// Elementwise helpers
// ---------------------------------------------------------------------------
__global__ void fill_f32_kernel(float* __restrict__ p, float v, int n) {
  const int i = blockIdx.x * blockDim.x + threadIdx.x;
  if (i < n) p[i] = v;
}

__global__ void zero4_kernel(float4* __restrict__ p, int n4) {
  const int i = blockIdx.x * blockDim.x + threadIdx.x;
  if (i < n4) p[i] = make_float4(0.f, 0.f, 0.f, 0.f);
}

// deg[dst[e]] += 1 (deg pre-filled with 1.0 to account for self-loops)
__global__ void degree_kernel(const int* __restrict__ dst,
                              float* __restrict__ deg, int E) {
  const int e = blockIdx.x * blockDim.x + threadIdx.x;
  if (e < E) atomicAdd(&deg[dst[e]], 1.0f);
}

__global__ void rsqrt_kernel(float* __restrict__ deg, int n) {
  const int i = blockIdx.x * blockDim.x + threadIdx.x;
  if (i < n) {
    const float d = deg[i];
    deg[i] = d > 0.f ? rsqrtf(d) : 0.f;
  }
}

// ---------------------------------------------------------------------------
// Edge aggregation: out[dst] += dis[src]*dis[dst] * h[src]
// D=128: one wave per edge (32 lanes x float4 = full 512B row, coalesced).
// ---------------------------------------------------------------------------
template <int D>
__global__ void scatter_edges_kernel(const float* __restrict__ h,
                                     const int* __restrict__ src,
                                     const int* __restrict__ dst,
                                     const float* __restrict__ dis,
                                     float* __restrict__ out, int E) {
  constexpr int C = D / 4;
  const int tid = blockIdx.x * blockDim.x + threadIdx.x;
  const int e = tid / C;
  const int c = tid % C;
  if (e >= E) return;
  const int s = src[e];
  const int d = dst[e];
  const float coef = dis[s] * dis[d];
  const float4 v = *(const float4*)(h + (size_t)s * D + 4 * c);
  float* o = out + (size_t)d * D + 4 * c;
  atomicAdd(o + 0, coef * v.x);
  atomicAdd(o + 1, coef * v.y);
  atomicAdd(o + 2, coef * v.z);
  atomicAdd(o + 3, coef * v.w);
}

// out[i] = relu(out[i] + dis[i]^2 * h[i] + bias)   (self-loop + bias + ReLU)
template <int D>
__global__ void finish_layer_kernel(const float* __restrict__ h,
                                    const float* __restrict__ dis,
                                    const float* __restrict__ bias,
                                    float* __restrict__ out, int M) {
  constexpr int C = D / 4;
  const int tid = blockIdx.x * blockDim.x + threadIdx.x;
  const int i = tid / C;
  const int c = tid % C;
  if (i >= M) return;
  const float coef = dis[i] * dis[i];
  const float4 v = *(const float4*)(h + (size_t)i * D + 4 * c);
  const float4 b = *(const float4*)(bias + 4 * c);
  float4 o = *(float4*)(out + (size_t)i * D + 4 * c);
  o.x = fmaxf(o.x + coef * v.x + b.x, 0.f);
  o.y = fmaxf(o.y + coef * v.y + b.y, 0.f);
  o.z = fmaxf(o.z + coef * v.z + b.z, 0.f);
  o.w = fmaxf(o.w + coef * v.w + b.w, 0.f);
  *(float4*)(out + (size_t)i * D + 4 * c) = o;
}

// ---------------------------------------------------------------------------
// Row L2-normalize (D=64): one wave per row, shuffle reduction (wave32).
// ---------------------------------------------------------------------------
__global__ __launch_bounds__(256) void normalize_rows_kernel(
    float* __restrict__ p, int M) {
  const int row = blockIdx.x * (blockDim.x >> 5) + (threadIdx.x >> 5);
  if (row >= M) return;
  const int lane = threadIdx.x & 31;
  float* r = p + (size_t)row * 64;
  float2 v = *(float2*)(r + 2 * lane);
  float ss = v.x * v.x + v.y * v.y;
#pragma unroll
  for (int off = 16; off > 0; off >>= 1) ss += __shfl_xor(ss, off, 32);
  const float scale = 1.0f / fmaxf(sqrtf(ss), 1e-12f);
  v.x *= scale;
  v.y *= scale;
  *(float2*)(r + 2 * lane) = v;
}

// ---------------------------------------------------------------------------
static inline int ceil_div(long long a, long long b) {
  return (int)((a + b - 1) / b);
}

extern "C" void kernel_launch(void* const* d_in, const int* in_sizes, int n_in,
                              void* d_out, int out_size, void* d_ws,
                              size_t ws_size, hipStream_t stream) {
  const float* x  = (const float*)d_in[0];
  const int*   ei = (const int*)d_in[1];
  const float* W1 = (const float*)d_in[2];
  const float* b1 = (const float*)d_in[3];
  const float* W2 = (const float*)d_in[4];
  const float* b2 = (const float*)d_in[5];
  const float* W3 = (const float*)d_in[6];
  const float* b3 = (const float*)d_in[7];
  const float* Wp = (const float*)d_in[8];

  const int M = in_sizes[0] / 128;  // 50000 nodes
  const int E = in_sizes[1] / 2;    // 800000 edges
  const int* src = ei;
  const int* dst = ei + E;
  float* out = (float*)d_out;

  float* bufA = (float*)d_ws;                 // M x 128
  float* bufB = bufA + (size_t)M * 128;       // M x 128
  float* dis  = bufB + (size_t)M * 128;       // M

  const int TPB = 256;

  // --- degree and d^-1/2 (self-loop folded in via 1.0 init) ---
  fill_f32_kernel<<<ceil_div(M, TPB), TPB, 0, stream>>>(dis, 1.0f, M);
  degree_kernel<<<ceil_div(E, TPB), TPB, 0, stream>>>(dst, dis, E);
  rsqrt_kernel<<<ceil_div(M, TPB), TPB, 0, stream>>>(dis, M);

  const int MT = ceil_div(M, 16);
  auto gemm_blocks = [&](int NT) { return ceil_div((long long)MT * NT, 8); };

  // --- layer 1: h1 = relu(agg(x @ W1) + b1) ---
  gemm_wmma_f32_kernel<128, 128, false>
      <<<gemm_blocks(8), TPB, 0, stream>>>(x, W1, bufA, M);
  zero4_kernel<<<ceil_div((long long)M * 32, TPB), TPB, 0, stream>>>(
      (float4*)bufB, M * 32);
  scatter_edges_kernel<128><<<ceil_div((long long)E * 32, TPB), TPB, 0, stream>>>(
      bufA, src, dst, dis, bufB, E);
  finish_layer_kernel<128><<<ceil_div((long long)M * 32, TPB), TPB, 0, stream>>>(
      bufA, dis, b1, bufB, M);

  // --- layer 2: h2 = relu(agg(h1 @ W2) + b2) ---
  gemm_wmma_f32_kernel<128, 128, false>
      <<<gemm_blocks(8), TPB, 0, stream>>>(bufB, W2, bufA, M);
  zero4_kernel<<<ceil_div((long long)M * 32, TPB), TPB, 0, stream>>>(
      (float4*)bufB, M * 32);
  scatter_edges_kernel<128><<<ceil_div((long long)E * 32, TPB), TPB, 0, stream>>>(
      bufA, src, dst, dis, bufB, E);
  finish_layer_kernel<128><<<ceil_div((long long)M * 32, TPB), TPB, 0, stream>>>(
      bufA, dis, b2, bufB, M);

  // --- layer 3: h3 = relu(agg(h2 @ W3) + b3), dim 64 ---
  gemm_wmma_f32_kernel<128, 64, false>
      <<<gemm_blocks(4), TPB, 0, stream>>>(bufB, W3, bufA, M);
  zero4_kernel<<<ceil_div((long long)M * 16, TPB), TPB, 0, stream>>>(
      (float4*)bufB, M * 16);
  scatter_edges_kernel<64><<<ceil_div((long long)E * 16, TPB), TPB, 0, stream>>>(
      bufA, src, dst, dis, bufB, E);
  finish_layer_kernel<64><<<ceil_div((long long)M * 16, TPB), TPB, 0, stream>>>(
      bufA, dis, b3, bufB, M);

  // --- projection: p = relu(h3 @ Wp), then row L2-normalize in place ---
  gemm_wmma_f32_kernel<64, 64, true>
      <<<gemm_blocks(4), TPB, 0, stream>>>(bufB, Wp, out, M);
  normalize_rows_kernel<<<ceil_div(M, 8), TPB, 0, stream>>>(out, M);
}